// MHCCoefficients_79353815761053
// MI455X (gfx1250) — compile-verified
//
#include <hip/hip_runtime.h>
#include <math.h>
#include <stdint.h>

typedef __attribute__((ext_vector_type(2))) float v2f;
typedef __attribute__((ext_vector_type(8))) float v8f;
typedef __attribute__((ext_vector_type(4))) unsigned int u32x4;
typedef __attribute__((ext_vector_type(8))) int i32x8;
typedef __attribute__((ext_vector_type(4))) int i32x4;

#define NC 1024
#define OUTD 24
#define NWAVES 4
#define TOKS_PER_BLOCK (NWAVES * 16)
#define KC 64               // K-chunk per TDM tile (floats)
#define NCHUNK (NC / KC)    // 16 chunks
#define ROWP (KC + 4)       // padded LDS row pitch (bank-conflict-free)

// ---------------------------------------------------------------------------
// Precompute Wg padded to 32 cols and packed as K-pairs so B-fragment loads
// are a single coalesced b64 per lane:
//   P[p*32 + n] = ( g[2p]*W[2p][n], g[2p+1]*W[2p+1][n] )   (n>=24 -> 0)
// Size: 512 * 32 * 8B = 128 KB (lives in d_ws); L2/WGP$-resident thereafter.
// ---------------------------------------------------------------------------
__global__ void prep_w_kernel(const float* __restrict__ g,
                              const float* __restrict__ W,
                              float2* __restrict__ P) {
  int idx = blockIdx.x * blockDim.x + threadIdx.x;
  if (idx >= 512 * 32) return;
  int p = idx >> 5;
  int n = idx & 31;
  int k0 = p * 2;
  float2 v;
  v.x = (n < OUTD) ? g[k0]     * W[k0 * OUTD + n]       : 0.0f;
  v.y = (n < OUTD) ? g[k0 + 1] * W[(k0 + 1) * OUTD + n] : 0.0f;
  P[idx] = v;
}

__device__ __forceinline__ float sigmoidf_(float z) {
  return 1.0f / (1.0f + __expf(-z));
}

// ---------------------------------------------------------------------------
// Tensor Data Mover: fetch a 16-row x KC-float 2D tile (row pitch NC floats in
// global) into LDS at lds_off, inserting 4 dwords of padding every KC dwords
// so the LDS row pitch is ROWP floats (conflict-free ds_load_b64 fragments).
// D# per CDNA5 ISA (2D: groups 2/3 zero / disabled).
// ---------------------------------------------------------------------------
__device__ __forceinline__ void tdm_load_tile(const float* gsrc, unsigned lds_off) {
  uint64_t va = (uint64_t)(uintptr_t)gsrc;
  u32x4 g0;
  g0[0] = 1u;                                             // count=1, user mode
  g0[1] = lds_off;                                        // lds_addr
  g0[2] = (unsigned)(va & 0xFFFFFFFFu);                   // global_addr[31:0]
  g0[3] = (unsigned)((va >> 32) & 0x01FFFFFFu) | (2u << 30); // addr[56:32] | type=2

  i32x8 g1;
  // data_size=2 (4B) | pad_enable | pad_interval=5 (64 dwords) | pad_amount=3 (4 dwords)
  g1[0] = (int)((2u << 16) | (1u << 20) | (5u << 22) | (3u << 25));
  g1[1] = (int)((unsigned)KC << 16);   // tensor_dim0 = KC  (bits 63:48)
  g1[2] = (int)(16u << 16);            // tensor_dim1 = 16  (bits 95:80)
  g1[3] = (int)((unsigned)KC << 16);   // tile_dim0  = KC   (bits 127:112)
  g1[4] = 16;                          // tile_dim1  = 16   (bits 143:128)
  g1[5] = NC;                          // tensor_dim0_stride = 1024 (bits 191:160)
  g1[6] = 0;
  g1[7] = 0;

  i32x4 z4 = {0, 0, 0, 0};
  i32x8 z8 = {0, 0, 0, 0, 0, 0, 0, 0};
  // 6-arg toolchain form: (g0, g1, g2, g3, g_extra, cpol)
  __builtin_amdgcn_tensor_load_to_lds(g0, g1, z4, z4, z8, 0);
}

// ---------------------------------------------------------------------------
// Fused: rmsnorm (g folded into W) + projection GEMM (fp32 WMMA, A staged via
// TDM into LDS) + sigmoid heads + 4x4 Sinkhorn-Knopp epilogue.
// One wave = 16 tokens (one WMMA M tile).
// ---------------------------------------------------------------------------
__global__ void __launch_bounds__(128)
mhc_kernel(const float* __restrict__ x, const float2* __restrict__ P,
           const float* __restrict__ alpha_pre, const float* __restrict__ alpha_post,
           const float* __restrict__ alpha_res, const float* __restrict__ b_pre,
           const float* __restrict__ b_post, const float* __restrict__ b_res,
           float* __restrict__ out, long total_tokens) {
  __shared__ __align__(16) float xs[NWAVES][2][16 * ROWP];   // TDM double buffers
  __shared__ __align__(16) float projs[NWAVES][16][OUTD];    // epilogue staging

  const int tid  = threadIdx.x;
  const int wave = tid >> 5;
  const int lane = tid & 31;
  const int m    = lane & 15;   // token row within tile (A-matrix M = lane%16)
  const int h    = lane >> 4;   // K-half select (ISA 16x4 f32 A layout)

  long tile0 = (long)blockIdx.x * TOKS_PER_BLOCK + (long)wave * 16;
  long tbase = tile0;
  if (tbase > total_tokens - 16) tbase = total_tokens - 16;  // clamp tail tile
  if (tbase < 0) tbase = 0;
  const float* __restrict__ rowbase = x + tbase * NC;

  const unsigned lds0 = (unsigned)(uintptr_t)&xs[wave][0][0];
  const unsigned lds1 = (unsigned)(uintptr_t)&xs[wave][1][0];

  v8f acc0 = {0.f, 0.f, 0.f, 0.f, 0.f, 0.f, 0.f, 0.f};  // cols 0..15
  v8f acc1 = {0.f, 0.f, 0.f, 0.f, 0.f, 0.f, 0.f, 0.f};  // cols 16..23 (+pad)
  float sumsq = 0.0f;

  const v2f* __restrict__ Pv = (const v2f*)P;

  // Prime the pipeline: chunk 0 -> buffer 0.
  tdm_load_tile(rowbase, lds0);

  for (int c = 0; c < NCHUNK; ++c) {
    if (c + 1 < NCHUNK) {
      // Buffer (c+1)&1 was last read in the previous iteration; make sure those
      // ds_loads have drained before the TDM overwrites it.
      asm volatile("s_wait_dscnt 0x0" ::: "memory");
      tdm_load_tile(rowbase + (c + 1) * KC, ((c + 1) & 1) ? lds1 : lds0);
      __builtin_amdgcn_s_wait_tensorcnt(1);  // chunk c complete (in-order)
    } else {
      __builtin_amdgcn_s_wait_tensorcnt(0);
    }

    const float* __restrict__ buf  = &xs[wave][c & 1][0];
    const float* __restrict__ arow = buf + m * ROWP + 2 * h;

#pragma unroll
    for (int kk = 0; kk < KC; kk += 4) {
      // A fragment from LDS: lane holds x[m][kg+2h], x[m][kg+2h+1]
      v2f a = *(const v2f*)(arow + kk);
      // B fragments from the packed weight table (WGP$/L2 resident)
      const int kg = c * KC + kk;
      const v2f* pb = Pv + (long)((kg >> 1) + h) * 32;
      v2f b0 = pb[m];
      v2f b1 = pb[m + 16];

      acc0 = __builtin_amdgcn_wmma_f32_16x16x4_f32(false, a, false, b0,
                                                   (short)0, acc0, false, false);
      acc1 = __builtin_amdgcn_wmma_f32_16x16x4_f32(false, a, false, b1,
                                                   (short)0, acc1, false, false);
      // fused sum-of-squares for rmsnorm (this lane's half of the channels)
      sumsq = fmaf(a.x, a.x, sumsq);
      sumsq = fmaf(a.y, a.y, sumsq);
    }
  }

  // Combine the two half-row partials (lanes l and l^16 share token m).
  float tot = sumsq + __shfl_xor(sumsq, 16, 32);
  float rms = rsqrtf(tot * (1.0f / NC) + 1e-6f);

  // Scatter D tiles to LDS. C/D layout: lanes 0-15 col=lane, rows j (M=j);
  // lanes 16-31 same cols, rows j+8.
  {
    float* pw = &projs[wave][0][0];
#pragma unroll
    for (int j = 0; j < 8; ++j) {
      int r = j + 8 * h;
      pw[r * OUTD + m] = acc0[j];
      if (m < 8) pw[r * OUTD + 16 + m] = acc1[j];
    }
  }
  __syncthreads();

  // Per-token epilogue: lane m (0..15) owns token tbase+m.
  const long tokm = tbase + m;
  if (lane < 16 && tile0 + m < total_tokens) {
    float pr[24];
    const float4* src = (const float4*)&projs[wave][m][0];  // 24 floats, 16B-aligned
#pragma unroll
    for (int q = 0; q < 6; ++q) {
      float4 v = src[q];
      pr[q * 4 + 0] = v.x * rms;
      pr[q * 4 + 1] = v.y * rms;
      pr[q * 4 + 2] = v.z * rms;
      pr[q * 4 + 3] = v.w * rms;
    }

    const float ap  = alpha_pre[0];
    const float apo = alpha_post[0];
    const float ar  = alpha_res[0];

    float hpre[4], hpost[4];
#pragma unroll
    for (int i = 0; i < 4; ++i) {
      hpre[i]  = sigmoidf_(fmaf(ap,  pr[i],     b_pre[i]));
      hpost[i] = sigmoidf_(fmaf(apo, pr[4 + i], b_post[i])) * 0.2f;  // (2/K)*0.1
    }

    // Sinkhorn-Knopp on 4x4: X = exp(alpha*dyn_res + b_res)
    float X[16];
#pragma unroll
    for (int i = 0; i < 16; ++i)
      X[i] = __expf(fmaf(ar, pr[8 + i], b_res[i]));

    float rr[4];
    float cc[4] = {1.f, 1.f, 1.f, 1.f};
#pragma unroll 1
    for (int t = 0; t < 20; ++t) {
#pragma unroll
      for (int i = 0; i < 4; ++i)
        rr[i] = 1.0f / (X[i * 4 + 0] * cc[0] + X[i * 4 + 1] * cc[1] +
                        X[i * 4 + 2] * cc[2] + X[i * 4 + 3] * cc[3]);
#pragma unroll
      for (int j = 0; j < 4; ++j)
        cc[j] = 1.0f / (X[0 * 4 + j] * rr[0] + X[1 * 4 + j] * rr[1] +
                        X[2 * 4 + j] * rr[2] + X[3 * 4 + j] * rr[3]);
    }

    float* out_pre  = out;
    float* out_post = out + 4 * total_tokens;
    float* out_res  = out + 8 * total_tokens;

    *(float4*)(out_pre  + tokm * 4) = make_float4(hpre[0], hpre[1], hpre[2], hpre[3]);
    *(float4*)(out_post + tokm * 4) = make_float4(hpost[0], hpost[1], hpost[2], hpost[3]);
#pragma unroll
    for (int i = 0; i < 4; ++i) {
      *(float4*)(out_res + tokm * 16 + i * 4) =
          make_float4(rr[i] * X[i * 4 + 0] * cc[0], rr[i] * X[i * 4 + 1] * cc[1],
                      rr[i] * X[i * 4 + 2] * cc[2], rr[i] * X[i * 4 + 3] * cc[3]);
    }
  }
}

extern "C" void kernel_launch(void* const* d_in, const int* in_sizes, int n_in,
                              void* d_out, int out_size, void* d_ws, size_t ws_size,
                              hipStream_t stream) {
  const float* x     = (const float*)d_in[0];
  const float* g     = (const float*)d_in[1];
  const float* W     = (const float*)d_in[2];
  const float* a_pre = (const float*)d_in[3];
  const float* a_pos = (const float*)d_in[4];
  const float* a_res = (const float*)d_in[5];
  const float* b_pre = (const float*)d_in[6];
  const float* b_pos = (const float*)d_in[7];
  const float* b_res = (const float*)d_in[8];

  const long T = (long)in_sizes[0] / NC;  // total tokens (B*S)

  // Workspace: 512*32 float2 = 128 KB packed, g-folded, zero-padded weights.
  float2* P = (float2*)d_ws;
  prep_w_kernel<<<(512 * 32 + 255) / 256, 256, 0, stream>>>(g, W, P);

  const long nblocks = (T + TOKS_PER_BLOCK - 1) / TOKS_PER_BLOCK;
  mhc_kernel<<<dim3((unsigned)nblocks), 128, 0, stream>>>(
      x, P, a_pre, a_pos, a_res, b_pre, b_pos, b_res, (float*)d_out, T);
}